// DTWKERNEL_69080253989227
// MI455X (gfx1250) — compile-verified
//
#include <hip/hip_runtime.h>
#include <hip/hip_bf16.h>

#define K_LEN 1024
#define I_LEN 8192
#define DIM   128

typedef __attribute__((ext_vector_type(2))) float v2f;
typedef __attribute__((ext_vector_type(8))) float v8f;

#define F_INF __builtin_inff()

// ---------------------------------------------------------------------------
// Kernel 1: row squared-norms for kernel (K rows) and x (I rows).
// One wave32 per row: 4 fp32 per lane, shuffle reduction.
// ---------------------------------------------------------------------------
__global__ void dtw_norms_kernel(const float* __restrict__ Kr,
                                 const float* __restrict__ X,
                                 float* __restrict__ k2,
                                 float* __restrict__ x2) {
    const int lane = threadIdx.x & 31;
    const int wave = (blockIdx.x * blockDim.x + threadIdx.x) >> 5;
    if (wave >= K_LEN + I_LEN) return;

    const float* row;
    float* outp;
    if (wave < K_LEN) { row = Kr + (size_t)wave * DIM;            outp = k2 + wave; }
    else              { row = X  + (size_t)(wave - K_LEN) * DIM;  outp = x2 + (wave - K_LEN); }

    float s = 0.0f;
#pragma unroll
    for (int t = lane; t < DIM; t += 32) {
        float v = row[t];
        s = fmaf(v, v, s);
    }
#pragma unroll
    for (int off = 16; off; off >>= 1)
        s += __shfl_xor(s, off, 32);
    if (lane == 0) *outp = s;
}

// ---------------------------------------------------------------------------
// Kernel 2: cost matrix C = max(k2[m] + x2[n] - 2*(Kr @ X^T), 0)
// via V_WMMA_F32_16X16X4_F32 (fp32 matrix core, exact accumulation).
// One 16x16 output tile per wave; 8 waves per block.
//
// ISA fragment layouts (CDNA5 ISA):
//   A 16x4 f32 : lane l holds row m = l%16 ; VGPR v holds K = v + 2*(l/16)
//   B 4x16 f32 : lane l holds col n = l%16 ; VGPR v holds K = v + 2*(l/16)
//   C/D 16x16  : lane l, VGPR v -> m = v + 8*(l/16), n = l%16
// ---------------------------------------------------------------------------
__global__ void dtw_cost_wmma_kernel(const float* __restrict__ Kr,
                                     const float* __restrict__ X,
                                     const float* __restrict__ k2,
                                     const float* __restrict__ x2,
                                     float* __restrict__ C) {
    const int lane  = threadIdx.x & 31;
    const int wave  = threadIdx.x >> 5;
    const int wpb   = blockDim.x >> 5;
    const int tile  = blockIdx.x * wpb + wave;
    const int tiles_n = I_LEN / 16;                 // 512
    if (tile >= (K_LEN / 16) * tiles_n) return;

    const int m0 = (tile / tiles_n) * 16;
    const int n0 = (tile % tiles_n) * 16;

    const int r    = lane & 15;                     // row (A) / col (B) in tile
    const int koff = 2 * (lane >> 4);               // per-lane K sub-offset

    const float* arow = Kr + (size_t)(m0 + r) * DIM + koff;
    const float* brow = X  + (size_t)(n0 + r) * DIM + koff;

    v8f acc = {};
#pragma unroll
    for (int kk = 0; kk < DIM; kk += 4) {
        v2f a, b;
        a[0] = arow[kk + 0];
        a[1] = arow[kk + 1];
        b[0] = brow[kk + 0];
        b[1] = brow[kk + 1];
        // D = A * B + C   (fp32 matrix op, 8-arg VOP3P form)
        acc = __builtin_amdgcn_wmma_f32_16x16x4_f32(
            /*neg_a=*/false, a, /*neg_b=*/false, b,
            /*c_mod=*/(short)0, acc, /*reuse_a=*/false, /*reuse_b=*/false);
    }

    const int n     = n0 + r;
    const int mbase = m0 + 8 * (lane >> 4);
    const float xn  = x2[n];
#pragma unroll
    for (int v = 0; v < 8; ++v) {
        const int m = mbase + v;
        float c = k2[m] + xn - 2.0f * acc[v];
        C[(size_t)m * I_LEN + n] = fmaxf(c, 0.0f);
    }
}

// ---------------------------------------------------------------------------
// Kernel 3: DTW DP wavefront. One workgroup, 1024 threads (32 waves),
// thread k owns row k. Anti-diagonal d: cell (k, i=d-k).
//   D[k,i] = c[k,i] + min(D[k-1,i-1], D[k-1,i], D[k,i-1])
// Neighbor-row values flow through a 3-deep LDS ring; own row's previous
// value stays in a register. Exactly one s_barrier per diagonal.
// ---------------------------------------------------------------------------
__global__ void dtw_dp_kernel(const float* __restrict__ C,
                              float* __restrict__ D) {
    __shared__ float ring[3][K_LEN];
    const int k = threadIdx.x;

    ring[0][k] = F_INF;
    ring[1][k] = F_INF;
    ring[2][k] = F_INF;
    float dleft = F_INF;                      // D[k][i-1]
    __syncthreads();

    const float* crow = C + (size_t)k * I_LEN;
    float* drow       = D + (size_t)k * I_LEN;

    for (int d = 0; d < K_LEN + I_LEN - 1; ++d) {
        const int i = d - k;
        float dcur = F_INF;
        if (i >= 0 && i < I_LEN) {
            if (i + 64 < I_LEN)
                __builtin_prefetch(&crow[i + 64], 0, 3);   // global_prefetch_b8
            const float c = crow[i];
            if ((k | i) == 0) {
                dcur = c;
            } else {
                // diag d-1 lives in ring[(d+2)%3], diag d-2 in ring[(d+1)%3]
                const float up   = (k > 0) ? ring[(d + 2) % 3][k - 1] : F_INF;
                const float diag = (k > 0) ? ring[(d + 1) % 3][k - 1] : F_INF;
                dcur = c + fminf(diag, fminf(up, dleft));
            }
            drow[i] = dcur;
        }
        dleft = dcur;                         // INF when cell invalid -> border
        ring[d % 3][k] = dcur;                // nobody reads slot d%3 this step
        __syncthreads();
    }
}

// ---------------------------------------------------------------------------
// Kernel 4: backtrack from (K-1, I-1) to (0,0), candidate order
// [diag, up, left] with first-tie (matches numpy.argmin), +inf border.
// Single wave32; the 128-dim squared distance of each emitted path cell is
// spread across lanes (4 MACs/lane/step); one shuffle reduction at the end.
// ---------------------------------------------------------------------------
__global__ void dtw_backtrack_kernel(const float* __restrict__ Kr,
                                     const float* __restrict__ X,
                                     const float* __restrict__ D,
                                     float* __restrict__ out) {
    const int lane = threadIdx.x & 31;
    int k = K_LEN - 1;
    int i = I_LEN - 1;
    bool fin = false;
    float acc = 0.0f;

    for (int step = 0; step < K_LEN + I_LEN - 1; ++step) {
        if (!fin) {
            const float* kr = Kr + (size_t)k * DIM;
            const float* xr = X  + (size_t)i * DIM;
#pragma unroll
            for (int t = lane; t < DIM; t += 32) {
                const float df = kr[t] - xr[t];
                acc = fmaf(df, df, acc);
            }
        }
        const bool fin2 = fin | ((k == 0) & (i == 0));

        const float diag = (k > 0 && i > 0) ? D[(size_t)(k - 1) * I_LEN + (i - 1)] : F_INF;
        const float up   = (k > 0)          ? D[(size_t)(k - 1) * I_LEN + i]       : F_INF;
        const float left = (i > 0)          ? D[(size_t)k * I_LEN + (i - 1)]       : F_INF;

        int a = 0;
        float best = diag;
        if (up   < best) { best = up;   a = 1; }
        if (left < best) { best = left; a = 2; }

        if (!fin2) {
            k -= (a != 2);
            i -= (a != 1);
        }
        fin = fin2;
    }

#pragma unroll
    for (int off = 16; off; off >>= 1)
        acc += __shfl_xor(acc, off, 32);
    if (lane == 0) out[0] = acc;
}

// ---------------------------------------------------------------------------
// Launch. Workspace layout (fp32):
//   [0, K*I)        : C cost matrix        (33.5 MB)
//   [K*I, 2*K*I)    : D accumulated matrix (33.5 MB)
//   [2*K*I, +K)     : k2 row norms
//   [.., +I)        : x2 row norms
// ---------------------------------------------------------------------------
extern "C" void kernel_launch(void* const* d_in, const int* in_sizes, int n_in,
                              void* d_out, int out_size, void* d_ws, size_t ws_size,
                              hipStream_t stream) {
    const float* kr = (const float*)d_in[0];   // (K_LEN, DIM) fp32
    const float* x  = (const float*)d_in[1];   // (I_LEN, DIM) fp32

    float* ws = (float*)d_ws;
    const size_t KI = (size_t)K_LEN * I_LEN;
    float* C  = ws;
    float* D  = ws + KI;
    float* k2 = ws + 2 * KI;
    float* x2 = k2 + K_LEN;

    // 1) norms: (K+I)=9216 waves, 8 waves / 256-thread block -> 1152 blocks
    dtw_norms_kernel<<<(K_LEN + I_LEN) / 8, 256, 0, stream>>>(kr, x, k2, x2);

    // 2) cost matrix: 64*512 = 32768 tiles, 8 waves/block -> 4096 blocks
    dtw_cost_wmma_kernel<<<(K_LEN / 16) * (I_LEN / 16) / 8, 256, 0, stream>>>(
        kr, x, k2, x2, C);

    // 3) DP wavefront: single 1024-thread workgroup (depth-optimal)
    dtw_dp_kernel<<<1, K_LEN, 0, stream>>>(C, D);

    // 4) backtrack + path sum: single wave
    dtw_backtrack_kernel<<<1, 32, 0, stream>>>(kr, x, D, (float*)d_out);
}